// ScanExample_66065186947681
// MI455X (gfx1250) — compile-verified
//
#include <hip/hip_runtime.h>

// out[d] = init[d] + sum_t xs[t][d],  xs: 131072 x 2048 f32.
// Pure HBM-bandwidth problem: 1 GiB read @ 23.3 TB/s => ~46 us floor.
// Stage 1: coalesced b128 streaming reduction into 256 partial rows (d_ws).
// Stage 2: WMMA_F32_16X16X4_F32 with a ones A-matrix reduces the partials.

#define T_ROWS 131072
#define D_COLS 2048
#define SLICES 256                      // partial rows in d_ws (2 MB)
#define ROWS_PER_SLICE (T_ROWS / SLICES) // 512

typedef __attribute__((ext_vector_type(2))) float v2f;
typedef __attribute__((ext_vector_type(8))) float v8f;

// ---------------- Stage 1: streaming partial column sums -------------------
// grid = (D/1024, SLICES), block = 256 threads (8 wave32s).
// Each thread owns one float4 column and ROWS_PER_SLICE rows:
// 512B per load per wave, 4 accumulators for ILP, prefetch 16 rows ahead.
__global__ void __launch_bounds__(256)
colsum_partial(const float* __restrict__ xs, float* __restrict__ partial) {
    const int stride4 = D_COLS / 4;                               // 512 float4/row
    const int c4      = blockIdx.x * blockDim.x + threadIdx.x;    // float4 column
    const long row0   = (long)blockIdx.y * ROWS_PER_SLICE;

    const float4* __restrict__ p = (const float4*)xs + row0 * stride4 + c4;

    float4 a0 = {0.f, 0.f, 0.f, 0.f};
    float4 a1 = {0.f, 0.f, 0.f, 0.f};
    float4 a2 = {0.f, 0.f, 0.f, 0.f};
    float4 a3 = {0.f, 0.f, 0.f, 0.f};

    for (int r = 0; r < ROWS_PER_SLICE; r += 4) {
        float4 v0 = p[0 * stride4];
        float4 v1 = p[1 * stride4];
        float4 v2 = p[2 * stride4];
        float4 v3 = p[3 * stride4];
        __builtin_prefetch(p + 16 * stride4, 0, 1);   // global_prefetch_b8
        a0.x += v0.x; a0.y += v0.y; a0.z += v0.z; a0.w += v0.w;
        a1.x += v1.x; a1.y += v1.y; a1.z += v1.z; a1.w += v1.w;
        a2.x += v2.x; a2.y += v2.y; a2.z += v2.z; a2.w += v2.w;
        a3.x += v3.x; a3.y += v3.y; a3.z += v3.z; a3.w += v3.w;
        p += 4 * stride4;
    }

    float4 s;
    s.x = (a0.x + a1.x) + (a2.x + a3.x);
    s.y = (a0.y + a1.y) + (a2.y + a3.y);
    s.z = (a0.z + a1.z) + (a2.z + a3.z);
    s.w = (a0.w + a1.w) + (a2.w + a3.w);
    ((float4*)partial)[(long)blockIdx.y * stride4 + c4] = s;
}

// ---------------- Stage 2: WMMA tree-reduction of the partials -------------
// One wave32 per 16-column tile. A = ones(16x4) so D = ones*B + C: every row
// of D is the column-sum of the 4x16 B tile. 64 chained WMMAs fold all 256
// partial rows; row 0 of the accumulator (VGPR0, lanes 0-15) is the answer.
//
// 32-bit B 4x16 lane layout (ISA 7.12.2): lanes 0-15 hold N=lane with
// K={0,1} in VGPR{0,1}; lanes 16-31 hold N=lane-16 with K={2,3}.
__global__ void __launch_bounds__(32)
colsum_final_wmma(const float* __restrict__ partial,
                  const float* __restrict__ init,
                  float* __restrict__ out) {
    const int lane = threadIdx.x;          // wave32
    const int col0 = blockIdx.x * 16;      // 128 column tiles
    const int n    = lane & 15;
    const int kb   = (lane >> 4) << 1;     // 0 or 2

    v2f a; a[0] = 1.0f; a[1] = 1.0f;       // ones A-matrix (16x4)
    v8f c = {};                            // f32 accumulator (16x16)

    const float* base = partial + col0 + n;
    for (int k = 0; k < SLICES; k += 4) {
        v2f b;
        b[0] = base[(long)(k + kb)     * D_COLS];
        b[1] = base[(long)(k + kb + 1) * D_COLS];
        // (neg_a, A, neg_b, B, c_mod, C, reuse_a, reuse_b)
        c = __builtin_amdgcn_wmma_f32_16x16x4_f32(
                false, a, false, b, (short)0, c, false, false);
    }

    if (lane < 16) {
        // D row M=0 lives in accumulator VGPR0, lanes 0-15 (N = lane).
        out[col0 + n] = c[0] + init[col0 + n];
    }
}

// ---------------------------------------------------------------------------
extern "C" void kernel_launch(void* const* d_in, const int* in_sizes, int n_in,
                              void* d_out, int out_size, void* d_ws, size_t ws_size,
                              hipStream_t stream) {
    const float* init = (const float*)d_in[0];   // (2048,)
    const float* xs   = (const float*)d_in[1];   // (131072, 2048)
    float* out        = (float*)d_out;           // (2048,)
    float* partial    = (float*)d_ws;            // SLICES*D_COLS*4 = 2 MB

    dim3 g1(D_COLS / (256 * 4), SLICES);         // (2, 256) blocks of 256
    colsum_partial<<<g1, 256, 0, stream>>>(xs, partial);

    colsum_final_wmma<<<D_COLS / 16, 32, 0, stream>>>(partial, init, out);
}